// HybridLoss_37297495998989
// MI455X (gfx1250) — compile-verified
//
#include <hip/hip_runtime.h>
#include <math.h>

// ---------------------------------------------------------------------------
// HybridLoss for MI455X (gfx1250, wave32):
//   focal(preds,targets) + contrastive(features,targets) + 0.1*graphreg(gfeat)
// Heavy part: features @ features^T (8192x8192x128) on the CDNA5 fp32 matrix
// path V_WMMA_F32_16X16X4_F32, with 2x4 register tiling (32x64 per wave) so
// LDS bandwidth (256 B/clk/WGP) is not the bottleneck, and async DMA staging
// (global_load_async_to_lds_b128 / s_wait_asynccnt) for memory->LDS tiles.
// ---------------------------------------------------------------------------

#define NROWS 8192
#define DIMS  128
#define PADW  132   // LDS row stride (floats): 132 mod 64 == 4 -> bank-conflict-free b64 reads

typedef float v2f __attribute__((ext_vector_type(2)));
typedef float v8f __attribute__((ext_vector_type(8)));

__device__ __forceinline__ float waveReduceSum(float v) {
#pragma unroll
    for (int m = 16; m >= 1; m >>= 1)
        v += __shfl_xor(v, m, 32);
    return v;
}

// LDS aperture flat pointers carry the wave-relative LDS byte offset in the
// low 32 bits.
__device__ __forceinline__ unsigned lds_off(const void* p) {
    return (unsigned)(unsigned long long)p;
}

// gfx1250 async DMA: memory -> LDS, 16B per lane, tracked by ASYNCcnt.
__device__ __forceinline__ void async_load_b128(unsigned ldsoff, const float* gaddr) {
    asm volatile("global_load_async_to_lds_b128 %0, %1, off"
                 :: "v"(ldsoff), "v"(gaddr) : "memory");
}

__device__ __forceinline__ void async_wait0() {
    asm volatile("s_wait_asynccnt 0" ::: "memory");
}

// ---------------- squared norms: one wave per row, coalesced float4 --------
__global__ void sqn_kernel(const float* __restrict__ feat, float* __restrict__ sqn) {
    const int wid  = (blockIdx.x * blockDim.x + threadIdx.x) >> 5;
    const int lane = threadIdx.x & 31;
    const float4 f = *(const float4*)(feat + (size_t)wid * DIMS + lane * 4);
    float s = f.x * f.x + f.y * f.y + f.z * f.z + f.w * f.w;
    s = waveReduceSum(s);
    if (lane == 0) sqn[wid] = s;
}

// ---------------- focal loss partials: 32 blocks x 256 threads -------------
__global__ void focal_kernel(const float* __restrict__ preds,
                             const float* __restrict__ targets,
                             float* __restrict__ partial) {
    const int i = blockIdx.x * blockDim.x + threadIdx.x;
    const float p = preds[i];
    const float t = targets[i];
    const float bce = fmaxf(p, 0.0f) - p * t + log1pf(expf(-fabsf(p)));
    const float pt  = expf(-bce);
    const float om  = 1.0f - pt;
    float v = 0.25f * om * om * bce;

    __shared__ float red[8];
    v = waveReduceSum(v);
    const int w = threadIdx.x >> 5;
    if ((threadIdx.x & 31) == 0) red[w] = v;
    __syncthreads();
    if (threadIdx.x == 0) {
        float s = 0.0f;
#pragma unroll
        for (int k = 0; k < 8; k++) s += red[k];
        partial[blockIdx.x] = s;
    }
}

// ---------------- graph regularizer partials: one wave per row diff --------
__global__ void graph_kernel(const float* __restrict__ g, float* __restrict__ partial) {
    const int wid  = (blockIdx.x * blockDim.x + threadIdx.x) >> 5;
    const int lane = threadIdx.x & 31;
    float nrm = 0.0f;
    if (wid < NROWS - 1) {                 // wave-uniform branch
        const float4 a = *(const float4*)(g + (size_t)wid * DIMS + lane * 4);
        const float4 b = *(const float4*)(g + (size_t)(wid + 1) * DIMS + lane * 4);
        const float dx = b.x - a.x, dy = b.y - a.y, dz = b.z - a.z, dw = b.w - a.w;
        float s = dx * dx + dy * dy + dz * dz + dw * dw;
        s = waveReduceSum(s);
        nrm = (lane == 0) ? sqrtf(s) : 0.0f;
    }
    __shared__ float red[8];
    const int w = threadIdx.x >> 5;
    if (lane == 0) red[w] = nrm;
    __syncthreads();
    if (threadIdx.x == 0) {
        float s = 0.0f;
#pragma unroll
        for (int k = 0; k < 8; k++) s += red[k];
        partial[blockIdx.x] = s;
    }
}

// ---------------- contrastive: 128x128 pair tile per block -----------------
// 8 waves as 4 row-slots x 2 col-slots, each wave owns a 32x64 tile
// (2x4 grid of 16x16 WMMA accumulators -> 8 WMMAs per 6 b64 LDS loads).
__global__ __launch_bounds__(256, 2)
void contrastive_kernel(const float* __restrict__ feat,
                        const float* __restrict__ targets,
                        const float* __restrict__ sqn,
                        float* __restrict__ partial) {
    __shared__ float As[128 * PADW];   // 128 feature rows (output rows)   ~66 KB
    __shared__ float Bs[128 * PADW];   // 128 feature rows (output cols)   ~66 KB
    __shared__ float sqA[128], tgA[128], sqB[128], tgB[128];
    __shared__ float red[8];

    const int rowBase = blockIdx.y * 128;
    const int colBase = blockIdx.x * 128;
    const int t = threadIdx.x;
    const int w = t >> 5, lane = t & 31;

    // Async-stage A and B tiles (DMA, no VGPR round trip). For iteration c,
    // wave w covers row (8c + w) fully: 512B contiguous per wave, coalesced.
#pragma unroll
    for (int c = 0; c < 16; c++) {
        const int chunk = t + c * 256;
        const int row = chunk >> 5;
        const int kk  = (chunk & 31) << 2;
        async_load_b128(lds_off(&As[row * PADW + kk]),
                        feat + (size_t)(rowBase + row) * DIMS + kk);
        async_load_b128(lds_off(&Bs[row * PADW + kk]),
                        feat + (size_t)(colBase + row) * DIMS + kk);
    }
    if (t < 128) { sqA[t] = sqn[rowBase + t];  tgA[t] = targets[rowBase + t]; }
    else         { const int j = t - 128;
                   sqB[j] = sqn[colBase + j];  tgB[j] = targets[colBase + j]; }
    async_wait0();
    __syncthreads();

    const int rowSlot = (w & 3) * 32;    // 4 row slots of 32
    const int colSlot = (w >> 2) * 64;   // 2 col slots of 64
    const int lrow  = lane & 15;
    const int khalf = (lane >> 4) << 1;  // lanes 16-31 hold K+2,K+3

    v8f acc[2][4];
#pragma unroll
    for (int ri = 0; ri < 2; ri++)
#pragma unroll
        for (int ci = 0; ci < 4; ci++)
            acc[ri][ci] = (v8f){0.f,0.f,0.f,0.f,0.f,0.f,0.f,0.f};

    const float* ap = &As[(rowSlot + lrow) * PADW + khalf];
    const float* bp = &Bs[(colSlot + lrow) * PADW + khalf];
#pragma unroll 2
    for (int k0 = 0; k0 < DIMS; k0 += 4) {
        v2f a[2], b[4];
#pragma unroll
        for (int ri = 0; ri < 2; ri++)
            a[ri] = *(const v2f*)(ap + ri * 16 * PADW + k0);
#pragma unroll
        for (int ci = 0; ci < 4; ci++)
            b[ci] = *(const v2f*)(bp + ci * 16 * PADW + k0);
#pragma unroll
        for (int ri = 0; ri < 2; ri++)
#pragma unroll
            for (int ci = 0; ci < 4; ci++)
                acc[ri][ci] = __builtin_amdgcn_wmma_f32_16x16x4_f32(
                    false, a[ri], false, b[ci],
                    (short)0, acc[ri][ci], false, false);
    }

    // Epilogue: C/D layout — lane L, vgpr v holds (M = v + 8*(L>=16), N = L&15).
    float sum = 0.0f;
    const int nIdx = lane & 15;
    const int mHi  = (lane >> 4) << 3;
#pragma unroll
    for (int ri = 0; ri < 2; ri++) {
#pragma unroll
        for (int ci = 0; ci < 4; ci++) {
            const int nn = colSlot + ci * 16 + nIdx;
            const float sqj = sqB[nn];
            const float tj  = tgB[nn];
#pragma unroll
            for (int v = 0; v < 8; v++) {
                const int m = rowSlot + ri * 16 + v + mHi;
                float sq = fmaxf(sqA[m] + sqj - 2.0f * acc[ri][ci][v], 0.0f);
                float per;
                if (tgA[m] == tj)    per = sq;           // same label: dist^2
                else if (sq < 1.0f) {                    // margin hit: rare
                    const float md = 1.0f - sqrtf(sq);
                    per = md * md;
                } else               per = 0.0f;         // skip sqrt entirely
                sum += per;
            }
        }
    }
    sum = waveReduceSum(sum);
    if (lane == 0) red[w] = sum;
    __syncthreads();
    if (t == 0) {
        float s = 0.0f;
#pragma unroll
        for (int k = 0; k < 8; k++) s += red[k];
        partial[blockIdx.y * gridDim.x + blockIdx.x] = s;
    }
}

// ---------------- final deterministic combine ------------------------------
__global__ void final_kernel(const float* __restrict__ ws, float* __restrict__ out) {
    const int t = threadIdx.x;   // 256 threads
    float f = 0.0f, g = 0.0f, c = 0.0f;
    if (t < 32) f = ws[t];
    for (int i = t; i < 1024; i += 256) g += ws[32 + i];
    for (int i = t; i < 4096; i += 256) c += ws[16384 + i];
    float v = f * (1.0f / 8192.0f)
            + c * (1.0f / 67108864.0f)          // / N^2
            + g * (0.1f / 8191.0f);             // LAMBDA * mean over N-1 diffs

    __shared__ float red[8];
    v = waveReduceSum(v);
    const int w = t >> 5;
    if ((t & 31) == 0) red[w] = v;
    __syncthreads();
    if (t == 0) {
        float s = 0.0f;
#pragma unroll
        for (int k = 0; k < 8; k++) s += red[k];
        out[0] = s;
    }
}

extern "C" void kernel_launch(void* const* d_in, const int* in_sizes, int n_in,
                              void* d_out, int out_size, void* d_ws, size_t ws_size,
                              hipStream_t stream) {
    const float* preds   = (const float*)d_in[0];
    const float* targets = (const float*)d_in[1];
    const float* feat    = (const float*)d_in[2];
    const float* gfeat   = (const float*)d_in[3];
    float* out = (float*)d_out;
    float* ws  = (float*)d_ws;

    // Workspace layout (floats):
    float* wsFocal = ws;            // [0,    32)
    float* wsGraph = ws + 32;       // [32,   1056)
    float* wsSqn   = ws + 2048;     // [2048, 10240)
    float* wsContr = ws + 16384;    // [16384,20480)  -> ~80 KB total

    sqn_kernel  <<<1024, 256, 0, stream>>>(feat, wsSqn);
    focal_kernel<<<  32, 256, 0, stream>>>(preds, targets, wsFocal);
    graph_kernel<<<1024, 256, 0, stream>>>(gfeat, wsGraph);

    dim3 grid(NROWS / 128, NROWS / 128);  // (64, 64) -> 4096 blocks
    contrastive_kernel<<<grid, 256, 0, stream>>>(feat, targets, wsSqn, wsContr);

    final_kernel<<<1, 256, 0, stream>>>(ws, out);
}